// RefillModelRNNConvolveSimple_78288663871813
// MI455X (gfx1250) — compile-verified
//
#include <hip/hip_runtime.h>
#include <hip/hip_bf16.h>
#include <math.h>

#define BB 8
#define LL 64
#define DD 256
#define VV 4096
#define MM 16
#define BL (BB*LL)

typedef __attribute__((ext_vector_type(2))) float v2f;
typedef __attribute__((ext_vector_type(8))) float v8f;

__device__ __forceinline__ v8f wmma4(v2f a, v2f b, v8f c) {
  // V_WMMA_F32_16X16X4_F32 : D = A(16x4 f32) * B(4x16 f32) + C(16x16 f32)
  return __builtin_amdgcn_wmma_f32_16x16x4_f32(
      /*neg_a=*/false, a, /*neg_b=*/false, b,
      /*c_mod=*/(short)0, c, /*reuse_a=*/false, /*reuse_b=*/false);
}

// ---------------- prep: yev = norm(embed[0]), xsa0v = norm(init_w[:,0]) --------
__global__ void prep_norm_kernel(const float* __restrict__ embed_w,
                                 const float* __restrict__ init_w,
                                 float* __restrict__ yev, float* __restrict__ xsa0v) {
  __shared__ float ss[256], sq[256];
  int tid = threadIdx.x;
  float e = embed_w[tid];                     // row 0 of embed
  ss[tid] = e; sq[tid] = e * e; __syncthreads();
  for (int s = 128; s > 0; s >>= 1) {
    if (tid < s) { ss[tid] += ss[tid + s]; sq[tid] += sq[tid + s]; }
    __syncthreads();
  }
  {
    float sum = ss[0], sumsq = sq[0];
    float var = (sumsq - sum * sum / (float)DD) / (float)(DD - 1);
    yev[tid] = e / (1e-5f + sqrtf(var));
  }
  __syncthreads();
  float i0 = init_w[tid * MM + 0];            // init_w[:,0], shape (D,M)
  ss[tid] = i0; sq[tid] = i0 * i0; __syncthreads();
  for (int s = 128; s > 0; s >>= 1) {
    if (tid < s) { ss[tid] += ss[tid + s]; sq[tid] += sq[tid + s]; }
    __syncthreads();
  }
  {
    float sum = ss[0], sumsq = sq[0];
    float var = (sumsq - sum * sum / (float)DD) / (float)(DD - 1);
    xsa0v[tid] = i0 / (1e-5f + sqrtf(var));
  }
}

// ---------------- ze[r] = norm(embed[z[r]]) ------------------------------------
__global__ void ze_kernel(const int* __restrict__ z, const float* __restrict__ embed_w,
                          float* __restrict__ ze) {
  __shared__ float ss[256], sq[256];
  int r = blockIdx.x, tid = threadIdx.x;
  float e = embed_w[(long)z[r] * DD + tid];
  ss[tid] = e; sq[tid] = e * e; __syncthreads();
  for (int s = 128; s > 0; s >>= 1) {
    if (tid < s) { ss[tid] += ss[tid + s]; sq[tid] += sq[tid + s]; }
    __syncthreads();
  }
  float sum = ss[0], sumsq = sq[0];
  float var = (sumsq - sum * sum / (float)DD) / (float)(DD - 1);
  ze[r * DD + tid] = e / (1e-5f + sqrtf(var));
}

// ---------------- kd = yev @ xkd_w^T + xkd_b ----------------------------------
__global__ void kd_kernel(const float* __restrict__ xkd_w, const float* __restrict__ xkd_b,
                          const float* __restrict__ yev, float* __restrict__ kd) {
  int d = threadIdx.x;
  float acc = 0.f;
  for (int j = 0; j < DD; ++j) acc += xkd_w[d * DD + j] * yev[j];
  kd[d] = acc + xkd_b[d];
}

// ---------------- broadcast xsa0v into xsaA -----------------------------------
__global__ void fill_xsa_kernel(const float* __restrict__ xsa0v, float* __restrict__ xsa) {
  int idx = blockIdx.x * 256 + threadIdx.x;
  xsa[idx] = xsa0v[idx & (DD - 1)];
}

// ---------------- generic WMMA GEMM: out = A @ W^T + bias  (BLxD)(DxD) --------
__global__ void gemmT_bias_kernel(const float* __restrict__ A, const float* __restrict__ W,
                                  const float* __restrict__ bias, float* __restrict__ out) {
  int wave = threadIdx.x >> 5, lane = threadIdx.x & 31;
  int tile = blockIdx.x * 8 + wave;
  const int NT = DD / 16;
  int tm = tile / NT, tn = tile % NT;
  int r0 = tm * 16, c0 = tn * 16;
  int m = lane & 15, half = lane >> 4, kb = half * 2, n = m;
  v8f acc = {};
  const float* arow = A + (long)(r0 + m) * DD;
  for (int k0 = 0; k0 < DD; k0 += 4) {
    v2f a, b;
    a.x = arow[k0 + kb];     a.y = arow[k0 + kb + 1];
    b.x = W[(c0 + n) * DD + k0 + kb];
    b.y = W[(c0 + n) * DD + k0 + kb + 1];
    acc = wmma4(a, b, acc);
  }
#pragma unroll
  for (int v = 0; v < 8; ++v) {
    int rr = r0 + v + 8 * half;
    out[rr * DD + c0 + n] = acc[v] + bias[c0 + n];
  }
}

// ---------------- one scan step ------------------------------------------------
// xnew[b,l] = (xsa[b,l-1]@Tw + xsa[b,l+1]@Tw^T + xsa[b,l]@Cw + cst[b,l]) / 3
__global__ void step_kernel(const float* __restrict__ xsa, const float* __restrict__ cst,
                            const float* __restrict__ trans_w, const float* __restrict__ core_w,
                            float* __restrict__ xnew) {
  int wave = threadIdx.x >> 5, lane = threadIdx.x & 31;
  int tile = blockIdx.x * 8 + wave;
  const int NT = DD / 16;
  int tm = tile / NT, tn = tile % NT;
  int r0 = tm * 16, c0 = tn * 16;
  int m = lane & 15, half = lane >> 4, kb = half * 2, n = m;
  int r = r0 + m;
  int b = r >> 6, l = r & (LL - 1);
  int rm1 = (b << 6) + ((l + LL - 1) & (LL - 1));  // roll(+1): take l-1
  int rp1 = (b << 6) + ((l + 1) & (LL - 1));       // roll(-1): take l+1
  v8f acc = {};
  const float* arow;
  // pass 1: A = xsa[l-1], B = trans_w (row-major)
  arow = xsa + (long)rm1 * DD;
  for (int k0 = 0; k0 < DD; k0 += 4) {
    v2f a, bf;
    a.x = arow[k0 + kb];  a.y = arow[k0 + kb + 1];
    bf.x = trans_w[(k0 + kb) * DD + c0 + n];
    bf.y = trans_w[(k0 + kb + 1) * DD + c0 + n];
    acc = wmma4(a, bf, acc);
  }
  // pass 2: A = xsa[l+1], B = trans_w^T
  arow = xsa + (long)rp1 * DD;
  for (int k0 = 0; k0 < DD; k0 += 4) {
    v2f a, bf;
    a.x = arow[k0 + kb];  a.y = arow[k0 + kb + 1];
    bf.x = trans_w[(c0 + n) * DD + k0 + kb];
    bf.y = trans_w[(c0 + n) * DD + k0 + kb + 1];
    acc = wmma4(a, bf, acc);
  }
  // pass 3: A = xsa[l], B = core_w (row-major)
  arow = xsa + (long)r * DD;
  for (int k0 = 0; k0 < DD; k0 += 4) {
    v2f a, bf;
    a.x = arow[k0 + kb];  a.y = arow[k0 + kb + 1];
    bf.x = core_w[(k0 + kb) * DD + c0 + n];
    bf.y = core_w[(k0 + kb + 1) * DD + c0 + n];
    acc = wmma4(a, bf, acc);
  }
#pragma unroll
  for (int v = 0; v < 8; ++v) {
    int rr = r0 + v + 8 * half;
    xnew[rr * DD + c0 + n] = (acc[v] + cst[rr * DD + c0 + n]) * (1.0f / 3.0f);
  }
}

// ------------- flash-softmax head: per row lse over V + logit at x[r] ---------
__global__ void head_lse_kernel(const float* __restrict__ Amat,   // BL x D
                                const float* __restrict__ embed,  // V x D
                                const int* __restrict__ x,        // BL
                                float* __restrict__ lse_out,      // BL
                                float* __restrict__ xlog_out) {   // BL
  __shared__ float As[16 * 257];
  __shared__ float smax[8][16], ssum[8][16];
  __shared__ float sxlog[16];
  __shared__ int sxt[16];
  int tid = threadIdx.x;
  int r0 = blockIdx.x * 16;
  for (int i = tid; i < 16 * DD; i += 256) {
    int row = i >> 8, k = i & 255;
    As[row * 257 + k] = Amat[(long)(r0 + row) * DD + k];
  }
  if (tid < 16) { sxt[tid] = x[r0 + tid]; sxlog[tid] = 0.f; }
  __syncthreads();

  int wave = tid >> 5, lane = tid & 31;
  int m = lane & 15, half = lane >> 4, kb = half * 2, n = m;
  float rmax[8], rsum[8];
#pragma unroll
  for (int v = 0; v < 8; ++v) { rmax[v] = -__builtin_inff(); rsum[v] = 0.f; }

  for (int t = wave; t < VV / 16; t += 8) {
    int v0 = t * 16;
    v8f acc = {};
    for (int k0 = 0; k0 < DD; k0 += 4) {
      v2f a, bf;
      a.x = As[m * 257 + k0 + kb];  a.y = As[m * 257 + k0 + kb + 1];
      bf.x = embed[(long)(v0 + n) * DD + k0 + kb];
      bf.y = embed[(long)(v0 + n) * DD + k0 + kb + 1];
      acc = wmma4(a, bf, acc);
    }
#pragma unroll
    for (int v = 0; v < 8; ++v) {
      int row = v + 8 * half;
      float val = acc[v];
      if (v0 + n == sxt[row]) sxlog[row] = val;      // exactly one writer per row
      float mx = val;
      mx = fmaxf(mx, __shfl_xor(mx, 1, 32));
      mx = fmaxf(mx, __shfl_xor(mx, 2, 32));
      mx = fmaxf(mx, __shfl_xor(mx, 4, 32));
      mx = fmaxf(mx, __shfl_xor(mx, 8, 32));
      float nm = fmaxf(rmax[v], mx);
      float ex = __expf(val - nm);
      ex += __shfl_xor(ex, 1, 32);
      ex += __shfl_xor(ex, 2, 32);
      ex += __shfl_xor(ex, 4, 32);
      ex += __shfl_xor(ex, 8, 32);
      rsum[v] = rsum[v] * __expf(rmax[v] - nm) + ex;
      rmax[v] = nm;
    }
  }
  if (m == 0) {
#pragma unroll
    for (int v = 0; v < 8; ++v) {
      int row = v + 8 * half;
      smax[wave][row] = rmax[v];
      ssum[wave][row] = rsum[v];
    }
  }
  __syncthreads();
  if (tid < 16) {
    float gm = -__builtin_inff();
    for (int w = 0; w < 8; ++w) gm = fmaxf(gm, smax[w][tid]);
    float gs = 0.f;
    for (int w = 0; w < 8; ++w) gs += ssum[w][tid] * __expf(smax[w][tid] - gm);
    lse_out[r0 + tid] = gm + logf(gs);
    xlog_out[r0 + tid] = sxlog[tid];
  }
}

// ---------------- yl[v] = yev . embed[v] --------------------------------------
__global__ void yl_kernel(const float* __restrict__ yev, const float* __restrict__ embed_w,
                          float* __restrict__ yl) {
  int v = blockIdx.x * 256 + threadIdx.x;
  float acc = 0.f;
  for (int j = 0; j < DD; ++j) acc += yev[j] * embed_w[(long)v * DD + j];
  yl[v] = acc;
}

__global__ void lsey_kernel(const float* __restrict__ yl, float* __restrict__ lseY) {
  __shared__ float sm[256];
  int tid = threadIdx.x;
  float mx = -__builtin_inff();
  for (int v = tid; v < VV; v += 256) mx = fmaxf(mx, yl[v]);
  sm[tid] = mx; __syncthreads();
  for (int s = 128; s > 0; s >>= 1) { if (tid < s) sm[tid] = fmaxf(sm[tid], sm[tid + s]); __syncthreads(); }
  float gmax = sm[0]; __syncthreads();
  float su = 0.f;
  for (int v = tid; v < VV; v += 256) su += __expf(yl[v] - gmax);
  sm[tid] = su; __syncthreads();
  for (int s = 128; s > 0; s >>= 1) { if (tid < s) sm[tid] += sm[tid + s]; __syncthreads(); }
  if (tid == 0) lseY[0] = gmax + logf(sm[0]);
}

// ---------------- per-row mixture combine -> cent[r] --------------------------
__global__ void final_kernel(const float* __restrict__ xsa, const float* __restrict__ xks_w,
                             const float* __restrict__ kd,
                             const float* __restrict__ lse0, const float* __restrict__ xl0,
                             const float* __restrict__ lse1, const float* __restrict__ xl1,
                             const float* __restrict__ yl, const float* __restrict__ lseY,
                             const int* __restrict__ x, float* __restrict__ cent) {
  __shared__ float s0s[256], s1s[256], s2s[256];
  int r = blockIdx.x, tid = threadIdx.x;
  float a = xsa[(long)r * DD + tid];
  s0s[tid] = a * xks_w[tid];
  s1s[tid] = a * xks_w[DD + tid];
  s2s[tid] = a * kd[tid];
  __syncthreads();
  for (int s = 128; s > 0; s >>= 1) {
    if (tid < s) { s0s[tid] += s0s[tid + s]; s1s[tid] += s1s[tid + s]; s2s[tid] += s2s[tid + s]; }
    __syncthreads();
  }
  if (tid == 0) {
    float s0 = s0s[0], s1 = s1s[0], s2 = s2s[0];
    float mm = fmaxf(s0, fmaxf(s1, s2));
    float e0 = __expf(s0 - mm), e1 = __expf(s1 - mm), e2 = __expf(s2 - mm);
    float Z = e0 + e1 + (float)LL * e2;
    float P0 = __expf(xl0[r] - lse0[r]);
    float P1 = __expf(xl1[r] - lse1[r]);
    float PY = __expf(yl[x[r]] - lseY[0]);
    cent[r] = sqrtf((e0 * P0 + e1 * P1 + (float)LL * e2 * PY) / Z);
  }
}

// ---------------- out[b] = -mean_l cent[b,l] ----------------------------------
__global__ void out_kernel(const float* __restrict__ cent, float* __restrict__ out) {
  __shared__ float s[64];
  int b = blockIdx.x, tid = threadIdx.x;
  s[tid] = cent[b * LL + tid]; __syncthreads();
  for (int st = 32; st > 0; st >>= 1) { if (tid < st) s[tid] += s[tid + st]; __syncthreads(); }
  if (tid == 0) out[b] = -s[0] / (float)LL;
}

extern "C" void kernel_launch(void* const* d_in, const int* in_sizes, int n_in,
                              void* d_out, int out_size, void* d_ws, size_t ws_size,
                              hipStream_t stream) {
  const int*   x       = (const int*)d_in[0];
  const int*   z       = (const int*)d_in[1];
  const float* embed_w = (const float*)d_in[2];
  const float* init_w  = (const float*)d_in[3];
  const float* trans_w = (const float*)d_in[4];
  const float* trans_b = (const float*)d_in[5];
  const float* core_w  = (const float*)d_in[6];
  const float* upd_w   = (const float*)d_in[7];
  const float* xks_w   = (const float*)d_in[8];
  const float* xkd_w   = (const float*)d_in[9];
  const float* xkd_b   = (const float*)d_in[10];
  const float* emi_w   = (const float*)d_in[11];
  const float* emi_b   = (const float*)d_in[12];
  float* out = (float*)d_out;
  float* ws  = (float*)d_ws;

  // workspace layout (floats)
  float* ze    = ws;                 // BL*D
  float* cst   = ze   + BL * DD;     // BL*D
  float* xsaA  = cst  + BL * DD;     // BL*D
  float* xsaB  = xsaA + BL * DD;     // BL*D
  float* emit  = xsaB + BL * DD;     // BL*D
  float* yev   = emit + BL * DD;     // D
  float* xsa0v = yev  + DD;          // D
  float* kd    = xsa0v + DD;         // D
  float* yl    = kd   + DD;          // V
  float* lse0  = yl   + VV;          // BL
  float* xl0   = lse0 + BL;          // BL
  float* lse1  = xl0  + BL;          // BL
  float* xl1   = lse1 + BL;          // BL
  float* lseY  = xl1  + BL;          // 1
  float* cent  = lseY + 1;           // BL

  // 1. normalized broadcast vectors
  prep_norm_kernel<<<1, 256, 0, stream>>>(embed_w, init_w, yev, xsa0v);
  // 2. kd and initial state
  kd_kernel<<<1, 256, 0, stream>>>(xkd_w, xkd_b, yev, kd);
  fill_xsa_kernel<<<BL * DD / 256, 256, 0, stream>>>(xsa0v, xsaA);
  // 3. ze = norm(embed[z])
  ze_kernel<<<BL, 256, 0, stream>>>(z, embed_w, ze);
  // 4. cst = ze @ upd_w^T + trans_b  (512 tiles / 8 waves)
  gemmT_bias_kernel<<<64, 256, 0, stream>>>(ze, upd_w, trans_b, cst);
  // 5. the scan: 64 serialized steps, ping-pong
  {
    const float* src = xsaA;
    float* dst = xsaB;
    for (int t = 0; t < LL; ++t) {
      step_kernel<<<64, 256, 0, stream>>>(src, cst, trans_w, core_w, dst);
      const float* ns = dst;
      dst = (float*)src;
      src = ns;
    }
    // LL even -> result back in xsaA
  }
  // 6. emit = xsa @ emi_w^T + emi_b
  gemmT_bias_kernel<<<64, 256, 0, stream>>>(xsaA, emi_w, emi_b, emit);
  // 7. flash softmax heads over V
  head_lse_kernel<<<BL / 16, 256, 0, stream>>>(emit, embed_w, x, lse0, xl0);
  head_lse_kernel<<<BL / 16, 256, 0, stream>>>(ze,   embed_w, x, lse1, xl1);
  // 8. yev-based softmax (single, shared by all rows)
  yl_kernel<<<VV / 256, 256, 0, stream>>>(yev, embed_w, yl);
  lsey_kernel<<<1, 256, 0, stream>>>(yl, lseY);
  // 9. combine mixture at target token
  final_kernel<<<BL, 256, 0, stream>>>(xsaA, xks_w, kd, lse0, xl0, lse1, xl1,
                                       yl, lseY, x, cent);
  // 10. reduce to (B,) output
  out_kernel<<<BB, 64, 0, stream>>>(cent, out);
}